// SoftSpearman_61641370632451
// MI455X (gfx1250) — compile-verified
//
#include <hip/hip_runtime.h>
#include <hip/hip_bf16.h>
#include <stdint.h>

typedef __attribute__((ext_vector_type(2))) float v2f;
typedef __attribute__((ext_vector_type(8))) float v8f;

#define N 2048      // vector length per row (reference V)
#define T 256       // threads per block = 8 wave32s
#define NW (T / 32)
#define CHUNKS (N / 4)      // K=4 per V_WMMA_F32_16X16X4_F32
#define CPW (CHUNKS / NW)   // wmma chunks per wave

__device__ __forceinline__ float logaddexpf_(float a, float b) {
  float m = fmaxf(a, b);
  float d = fminf(a, b) - m;
  return m + log1pf(__expf(d));
}

// Closed-form lse of the KL-PAV log-weights over block [i, kend]:
// w_j = log(N-j)  =>  lse = log( sum_{j=i..kend} (N-j) )  (triangular numbers).
// Replaces the reference's sequential logaddexp over lse_w exactly.
__device__ __forceinline__ float logW_(int i, int kend) {
  int a = N - i;
  int b = N - 1 - kend;
  int W = (a * (a + 1) - b * (b + 1)) >> 1;
  return __logf((float)W);
}

// CDNA5 async global->LDS copy, 16 bytes per lane, GV addressing.
// Tracked by ASYNCcnt (CDNA5 ISA).
__device__ __forceinline__ void async_cp16(uint32_t lds_off, const float* g) {
  asm volatile("global_load_async_to_lds_b128 %0, %1, off"
               :: "v"(lds_off), "v"(g)
               : "memory");
}

// Serial KL-isotonic PAV (torchsort isotonic_kl port) + block reconstruction.
// Runs on ONE thread; arrays live in LDS. lse_w handled in closed form.
__device__ void pav_kl_serial(float* ly, float* sol, unsigned short* tgt) {
  int i = 0;
  while (i < N) {
    int k = (int)tgt[i] + 1;
    if (k >= N) break;
    if (sol[i] > sol[k]) { i = k; continue; }
    float sol_prev = sol[i];
    for (;;) {
      float y = logaddexpf_(ly[i], ly[k]);
      int kend = (int)tgt[k];
      ly[i] = y;
      sol[i] = y - logW_(i, kend);
      int kn = kend + 1;
      if (kn >= N || sol_prev > sol[kn]) {
        tgt[i] = (unsigned short)(kn - 1);
        tgt[kn - 1] = (unsigned short)i;
        if (i > 0) i = (int)tgt[i - 1];
        break;
      }
      k = kn;
    }
  }
  // expand block heads into block-constant solution (reference lax.scan)
  int be = -1;
  float val = sol[0];
  for (int j = 0; j < N; ++j) {
    if (j > be) { val = sol[j]; be = (int)tgt[j]; }
    sol[j] = val;
  }
}

__global__ __launch_bounds__(T)
void soft_spearman_kernel(const float* __restrict__ x1,
                          const float* __restrict__ x2,
                          float* __restrict__ out) {
  // Exactly 64KB dynamic LDS => 5 resident blocks per 320KB WGP, maximizing
  // the number of concurrently running serial PAV chains.
  extern __shared__ char smem[];
  float* key  = (float*)smem;                    // [2N] theta, sorted desc
  float* lseY = key + 2 * N;                     // [2N] PAV lse_y -> ranks
  float* solA = lseY + 2 * N;                    // [2N] PAV dual solution
  unsigned short* idxA = (unsigned short*)(solA + 2 * N); // [2N] permutation
  unsigned short* tgtA = idxA + 2 * N;           // [2N] PAV block targets
  float* acc  = (float*)(void*)tgtA;             // [8] aliases tgt (dead then)

  const int tid = threadIdx.x;
  const int row = blockIdx.x;
  const float* s1 = x1 + (size_t)row * N;
  const float* s2 = x2 + (size_t)row * N;

  // ---- async stage both rows into LDS (ASYNCcnt path) ----
  for (int e = tid; e < N / 4; e += T) {
    async_cp16((uint32_t)(uintptr_t)&key[4 * e],     s1 + 4 * e);
    async_cp16((uint32_t)(uintptr_t)&key[N + 4 * e], s2 + 4 * e);
  }
  asm volatile("s_wait_asynccnt 0x0" ::: "memory");
  __syncthreads();

  // ---- theta = x / EPS ; init permutation ----
  for (int j = tid; j < 2 * N; j += T) {
    key[j] *= 10.0f;        // 1/EPS, EPS = 0.1
    idxA[j] = (unsigned short)(j & (N - 1));
  }

  // ---- bitonic sort (descending), both signals share one barrier schedule --
  for (int kk = 2; kk <= N; kk <<= 1) {
    for (int jj = kk >> 1; jj > 0; jj >>= 1) {
      __syncthreads();
      for (int i = tid; i < 2 * N; i += T) {
        int il = i & (N - 1);
        int base = i & ~(N - 1);
        int p = il ^ jj;
        if (p > il) {
          bool up = ((il & kk) == 0);
          float a = key[base + il], b = key[base + p];
          if (up ? (a < b) : (a > b)) {
            key[base + il] = b; key[base + p] = a;
            unsigned short t = idxA[base + il];
            idxA[base + il] = idxA[base + p];
            idxA[base + p] = t;
          }
        }
      }
    }
  }
  __syncthreads();

  // ---- PAV init: lse_y = s, sol = s - log(n-j), target = j ----
  for (int j = tid; j < 2 * N; j += T) {
    int jl = j & (N - 1);
    lseY[j] = key[j];
    solA[j] = key[j] - __logf((float)(N - jl));
    tgtA[j] = (unsigned short)jl;
  }
  __syncthreads();

  // ---- serial PAV, both signals concurrently on different waves ----
  if (tid == 0)  pav_kl_serial(lseY,     solA,     tgtA);
  if (tid == 32) pav_kl_serial(lseY + N, solA + N, tgtA + N);
  __syncthreads();

  // ---- ranks = exp(s - sol), scatter through perm (reuse lseY) ----
  // tgt region is dead now: zero the aliased moment accumulators.
  for (int j = tid; j < 2 * N; j += T) {
    int base = j & ~(N - 1);
    float r = __expf(key[j] - solA[j]);
    lseY[base + (int)idxA[j]] = r;
  }
  if (tid < 8) acc[tid] = 0.0f;
  __syncthreads();

  // ---- moment matrix via WMMA: D = M^T M, M rows = [r1; r2; 1] ----
  // Branchless tile build: one unconditional b64 LDS load per lane + selects,
  // so EXEC stays all-ones into v_wmma (required) and no exec ping-pong.
  // A layout (f32 16x4): lanes 0-15 -> M=lane, VGPR0=K0,VGPR1=K1;
  //                      lanes 16-31 -> M=lane-16, K2,K3. B mirrors by column.
  const int ln = tid & 31;
  const int wv = tid >> 5;
  const int m  = ln & 15;
  const int kb = (ln >> 4) * 2;
  const float* rsel = lseY + (m == 1 ? N : 0);   // row source (garbage ok m>=2)
  const float selv = (m < 2) ? 1.0f : 0.0f;
  const float addv = (m == 2) ? 1.0f : 0.0f;
  v8f c = {0.f, 0.f, 0.f, 0.f, 0.f, 0.f, 0.f, 0.f};
  for (int ch = wv * CPW; ch < wv * CPW + CPW; ++ch) {
    int p = ch * 4;
    float2 pr = *(const float2*)(rsel + p + kb);  // ds_load_b64, all lanes
    v2f ab;
    ab.x = pr.x * selv + addv;
    ab.y = pr.y * selv + addv;
    c = __builtin_amdgcn_wmma_f32_16x16x4_f32(
        /*neg_a=*/false, ab, /*neg_b=*/false, ab,
        /*c_mod=*/(short)0, c, /*reuse_a=*/false, /*reuse_b=*/false);
  }
  // D VGPR j: lanes0-15 -> M=j, N=lane.  D[0][0]=S11 D[0][1]=S12 D[1][1]=S22
  // D[0][2]=S1 D[1][2]=S2
  if (ln == 0)  atomicAdd(&acc[0], c[0]);                       // S11
  if (ln == 1) { atomicAdd(&acc[1], c[0]);                      // S12
                 atomicAdd(&acc[2], c[1]); }                    // S22
  if (ln == 2) { atomicAdd(&acc[3], c[0]);                      // S1
                 atomicAdd(&acc[4], c[1]); }                    // S2
  __syncthreads();

  if (tid == 0) {
    float S11 = acc[0], S12 = acc[1], S22 = acc[2], S1 = acc[3], S2 = acc[4];
    float nn = (float)N;
    float num = S12 - S1 * S2 / nn;
    float den = sqrtf((S11 - S1 * S1 / nn) * (S22 - S2 * S2 / nn));
    out[row] = 1.0f - num / den;
  }
}

extern "C" void kernel_launch(void* const* d_in, const int* in_sizes, int n_in,
                              void* d_out, int out_size, void* d_ws, size_t ws_size,
                              hipStream_t stream) {
  (void)n_in; (void)d_ws; (void)ws_size; (void)out_size;
  const float* x1 = (const float*)d_in[0];
  const float* x2 = (const float*)d_in[1];
  float* out = (float*)d_out;

  const int rows = in_sizes[0] / N;  // 16*256 = 4096
  // key(16K) + lseY(16K) + sol(16K) + idx(8K) + tgt(8K, aliased by acc) = 64KB
  const size_t smem = (size_t)(2 * N) * (3 * sizeof(float) + 2 * sizeof(unsigned short));

  hipFuncSetAttribute((const void*)soft_spearman_kernel,
                      hipFuncAttributeMaxDynamicSharedMemorySize, (int)smem);
  soft_spearman_kernel<<<rows, T, smem, stream>>>(x1, x2, out);
}